// DiscreteLaplacianLoss_69655779607181
// MI455X (gfx1250) — compile-verified
//
#include <hip/hip_runtime.h>
#include <hip/hip_bf16.h>

// CDNA5 / gfx1250 (wave32). D = 16 features.
#define DF 16

typedef __attribute__((ext_vector_type(2))) float v2f;
typedef __attribute__((ext_vector_type(8))) float v8f;

// ---------------------------------------------------------------------------
// Kernel 1: T = pred - inp ; S = 0 ; cnt = 0 ; out = 0
// ---------------------------------------------------------------------------
__global__ void dl_init(const float* __restrict__ pred,
                        const float* __restrict__ inp,
                        float* __restrict__ T,
                        float* __restrict__ S,
                        unsigned* __restrict__ C,
                        float* __restrict__ out,
                        int N) {
  int i = blockIdx.x * blockDim.x + threadIdx.x;
  int total = N * DF;
  if (i < total) {
    T[i] = pred[i] - inp[i];
    S[i] = 0.0f;
  }
  if (i < N) C[i] = 0u;
  if (i == 0) out[0] = 0.0f;
}

// ---------------------------------------------------------------------------
// Kernel 2: scatter pass. 16 lanes cooperate on one edge:
//   S[row][0..15] += T[col][0..15]  (coalesced 64B gather, 64B-local atomics)
//   C[row] += 1                     (lane d==0 only)
// Edge indices loaded by one lane per half-wave and broadcast via shuffle.
// ---------------------------------------------------------------------------
__global__ void dl_edges(const int* __restrict__ edges,   // [2, E] flat: row then col
                         const float* __restrict__ T,
                         float* __restrict__ S,
                         unsigned* __restrict__ C,
                         int E) {
  int i = blockIdx.x * blockDim.x + threadIdx.x;
  int lane = threadIdx.x & 31;
  int d = lane & 15;          // feature index
  int e = i >> 4;             // edge index (16 lanes per edge)
  bool valid = (e < E);
  int ec = valid ? e : (E - 1);

  int r = 0, c = 0;
  if (d == 0) {
    r = edges[ec];                 // row
    c = edges[(size_t)E + ec];     // col
    // sequential edge stream: hint the prefetcher ahead (global_prefetch_b8)
    __builtin_prefetch(edges + ec + 2048, 0, 0);
    __builtin_prefetch(edges + (size_t)E + ec + 2048, 0, 0);
  }
  int src = lane & 16;             // broadcast source: lane 0 or lane 16
  r = __shfl(r, src, 32);
  c = __shfl(c, src, 32);

  if (valid) {
    float v = T[(size_t)c * DF + d];                 // coalesced 64B line per edge
    atomicAdd(&S[(size_t)r * DF + d], v);            // global_atomic_add_f32 (no-rtn)
    if (d == 0) atomicAdd(&C[r], 1u);
  }
}

// ---------------------------------------------------------------------------
// Kernel 3: node pass. One wave owns a 16-node x 16-feature tile.
//   diff = keep*T - inv*S   (keep=0,inv=0 when cnt==0 or node out of range)
// Squared elementwise into the 16x4 f32 A-operand layout, multiplied by an
// all-ones 4x16 B with V_WMMA_F32_16X16X4_F32, accumulating the 4 K-chunks.
// The 16x16 accumulator then holds per-row sums replicated in every column;
// total tile sum = sum of acc[0..7] at lane 0 (rows 0-7) + lane 16 (rows 8-15).
// ---------------------------------------------------------------------------
__global__ void dl_nodes(const float* __restrict__ T,
                         const float* __restrict__ S,
                         const unsigned* __restrict__ C,
                         float* __restrict__ out,
                         int N, int tiles) {
  int lane = threadIdx.x & 31;
  int wave = (blockIdx.x * blockDim.x + threadIdx.x) >> 5;
  int nwaves = (gridDim.x * blockDim.x) >> 5;
  float invN = 1.0f / (float)N;

  v2f ones;
  ones.x = 1.0f;
  ones.y = 1.0f;

  for (int t = wave; t < tiles; t += nwaves) {
    int rowi = lane & 15;                 // matrix row == node within tile
    int node = t * 16 + rowi;
    bool ok = (node < N);
    int na = ok ? node : (N - 1);         // clamp address, mask value below
    int koff = (lane >> 4) << 1;          // lanes 0-15 hold K=k0,k0+1; 16-31 hold K=k0+2,k0+3

    unsigned cnt = C[na];
    bool live = ok && (cnt > 0u);
    float keep = live ? 1.0f : 0.0f;
    float inv = live ? (1.0f / (float)cnt) : 0.0f;

    const float* tb = T + (size_t)na * DF + koff;
    const float* sb = S + (size_t)na * DF + koff;

    v8f acc = {};
#pragma unroll
    for (int k0 = 0; k0 < DF; k0 += 4) {
      float2 tv = *(const float2*)(tb + k0);
      float2 sv = *(const float2*)(sb + k0);
      float d0 = keep * tv.x - inv * sv.x;
      float d1 = keep * tv.y - inv * sv.y;
      v2f a;
      a.x = d0 * d0;
      a.y = d1 * d1;
      // D(16x16,f32) += A(16x4,f32) x B(4x16,ones)
      acc = __builtin_amdgcn_wmma_f32_16x16x4_f32(
          /*neg_a=*/false, a, /*neg_b=*/false, ones,
          /*c_mod=*/(short)0, acc, /*reuse_a=*/false, /*reuse_b=*/false);
    }

    float s = acc[0] + acc[1] + acc[2] + acc[3] + acc[4] + acc[5] + acc[6] + acc[7];
    float tileSum = __shfl(s, 0, 32) + __shfl(s, 16, 32);
    if (lane == 0) atomicAdd(out, tileSum * invN);
  }
}

// ---------------------------------------------------------------------------
// Host launcher
// ---------------------------------------------------------------------------
extern "C" void kernel_launch(void* const* d_in, const int* in_sizes, int n_in,
                              void* d_out, int out_size, void* d_ws, size_t ws_size,
                              hipStream_t stream) {
  (void)n_in; (void)out_size; (void)ws_size;

  const float* pred = (const float*)d_in[0];   // [N,16] f32
  const float* inp  = (const float*)d_in[1];   // [N,16] f32
  const int* edges  = (const int*)d_in[2];     // [2,E] int

  const int N = in_sizes[0] / DF;              // 100000
  const int E = in_sizes[2] / 2;               // 3200000
  float* out = (float*)d_out;

  // Workspace layout: T [N*16 f32] | S [N*16 f32] | C [N u32]  (~13.2 MB)
  float* T = (float*)d_ws;
  float* S = T + (size_t)N * DF;
  unsigned* C = (unsigned*)(S + (size_t)N * DF);

  // Pass 1: init (covers N*16 elements; also zeros cnt and out)
  {
    int total = N * DF;
    int blocks = (total + 255) / 256;
    dl_init<<<blocks, 256, 0, stream>>>(pred, inp, T, S, C, out, N);
  }

  // Pass 2: edge scatter (16 lanes per edge)
  {
    long long total = (long long)E * DF;       // 51.2M lane-tasks
    int blocks = (int)((total + 255) / 256);
    dl_edges<<<blocks, 256, 0, stream>>>(edges, T, S, C, E);
  }

  // Pass 3: node reduction via WMMA (one wave per 16-node tile, grid-strided)
  {
    int tiles = (N + 15) / 16;                 // 6250
    int blocks = (tiles * 32 + 255) / 256;     // 1 wave per tile
    dl_nodes<<<blocks, 256, 0, stream>>>(T, S, C, out, N, tiles);
  }
}